// GCN_2_layer_fc_45311904973180
// MI455X (gfx1250) — compile-verified
//
#include <hip/hip_runtime.h>

typedef __attribute__((ext_vector_type(2))) float v2f;
typedef __attribute__((ext_vector_type(8))) float v8f;

#define NFEAT 128

// ---------------------------------------------------------------------------
// Degree count via float atomics (deg fits exactly in fp32: max 800000)
// ---------------------------------------------------------------------------
__global__ void deg_kernel(const long long* __restrict__ src,
                           const long long* __restrict__ dst,
                           float* __restrict__ deg_src,
                           float* __restrict__ deg_dst, int E) {
    int t = blockIdx.x * blockDim.x + threadIdx.x;
    if (t < E) {
        atomicAdd(&deg_src[(int)src[t]], 1.0f);
        atomicAdd(&deg_dst[(int)dst[t]], 1.0f);
    }
}

// deg -> (deg > 0 ? rsqrt(deg) : 0), applied to both arrays (contiguous 2N)
__global__ void norm_kernel(float* __restrict__ deg, int n2) {
    int t = blockIdx.x * blockDim.x + threadIdx.x;
    if (t < n2) {
        float d = deg[t];
        deg[t] = (d > 0.0f) ? rsqrtf(d) : 0.0f;
    }
}

// ---------------------------------------------------------------------------
// Y[M x FOUT] = opt_relu( (X[M x K] * rowscale[m]) @ W[K x FOUT] + bias )
// One wave32 computes a 32x32 output tile via 2x2 register blocking of
// V_WMMA_F32_16X16X4_F32 (4 WMMAs per K-step, A/B fragments reused).
// grid.x = ceil(M/32) ; blockDim.x = 32 * (FOUT/32).
// A-row loads clamp to M-1 (branchless: EXEC stays all-ones for WMMA);
// only the final stores are row-predicated.
// ---------------------------------------------------------------------------
template <int K, int FOUT, bool HAS_RS, bool HAS_BIAS, bool RELU>
__global__ void gemm_wmma_f32(const float* __restrict__ X,
                              const float* __restrict__ W,
                              const float* __restrict__ bias,
                              const float* __restrict__ rowscale,
                              float* __restrict__ Y,
                              int M) {
    const int lane = threadIdx.x & 31;
    const int wave = threadIdx.x >> 5;
    const int half = lane >> 4;        // 0: K sub 0..1, 1: K sub 2..3
    const int l    = lane & 15;
    const int row0 = blockIdx.x * 32;  // 32-row tile
    const int col0 = wave * 32;        // 32-col tile

    const int ar0 = min(row0 + l,      M - 1);   // clamped A rows
    const int ar1 = min(row0 + 16 + l, M - 1);
    const int n0  = col0 + l;                     // B/C/D columns
    const int n1  = col0 + 16 + l;

    const float rs0 = HAS_RS ? rowscale[ar0] : 1.0f;
    const float rs1 = HAS_RS ? rowscale[ar1] : 1.0f;

    const float* xr0 = X + (size_t)ar0 * K + half * 2;
    const float* xr1 = X + (size_t)ar1 * K + half * 2;
    const float* wp0 = W + (size_t)(half * 2) * FOUT + n0;
    const float* wp1 = W + (size_t)(half * 2) * FOUT + n1;

    v8f acc00 = {}, acc01 = {}, acc10 = {}, acc11 = {};

#pragma unroll 4
    for (int k0 = 0; k0 < K; k0 += 4) {
        v2f a0, a1, b0, b1;
        a0.x = xr0[k0] * rs0;  a0.y = xr0[k0 + 1] * rs0;
        a1.x = xr1[k0] * rs1;  a1.y = xr1[k0 + 1] * rs1;
        b0.x = wp0[k0 * FOUT]; b0.y = wp0[(k0 + 1) * FOUT];
        b1.x = wp1[k0 * FOUT]; b1.y = wp1[(k0 + 1) * FOUT];
        acc00 = __builtin_amdgcn_wmma_f32_16x16x4_f32(false, a0, false, b0, (short)0, acc00, false, false);
        acc01 = __builtin_amdgcn_wmma_f32_16x16x4_f32(false, a0, false, b1, (short)0, acc01, false, false);
        acc10 = __builtin_amdgcn_wmma_f32_16x16x4_f32(false, a1, false, b0, (short)0, acc10, false, false);
        acc11 = __builtin_amdgcn_wmma_f32_16x16x4_f32(false, a1, false, b1, (short)0, acc11, false, false);
    }

    const float bv0 = HAS_BIAS ? bias[n0] : 0.0f;
    const float bv1 = HAS_BIAS ? bias[n1] : 0.0f;

#pragma unroll
    for (int r = 0; r < 8; ++r) {
        // C/D layout: lane<16 -> M=r, lane>=16 -> M=r+8
        const int m0 = row0 + half * 8 + r;        // rows of acc0x
        const int m1 = row0 + 16 + half * 8 + r;   // rows of acc1x
        if (m0 < M) {
            float v00 = acc00[r] + bv0;
            float v01 = acc01[r] + bv1;
            if (RELU) { v00 = fmaxf(v00, 0.0f); v01 = fmaxf(v01, 0.0f); }
            Y[(size_t)m0 * FOUT + n0] = v00;
            Y[(size_t)m0 * FOUT + n1] = v01;
        }
        if (m1 < M) {
            float v10 = acc10[r] + bv0;
            float v11 = acc11[r] + bv1;
            if (RELU) { v10 = fmaxf(v10, 0.0f); v11 = fmaxf(v11, 0.0f); }
            Y[(size_t)m1 * FOUT + n0] = v10;
            Y[(size_t)m1 * FOUT + n1] = v11;
        }
    }
}

// ---------------------------------------------------------------------------
// agg[dst[e], :] += h[src[e], :]  — one wave per edge, 4 floats per lane
// ---------------------------------------------------------------------------
__global__ void scatter_kernel(const float* __restrict__ h,
                               const long long* __restrict__ src,
                               const long long* __restrict__ dst,
                               float* __restrict__ agg, int E) {
    long long t = (long long)blockIdx.x * blockDim.x + threadIdx.x;
    int e = (int)(t >> 5);
    int c = ((int)t & 31) * 4;
    if (e < E) {
        const float4 v = *(const float4*)(h + (size_t)src[e] * NFEAT + c);
        float* o = agg + (size_t)dst[e] * NFEAT + c;
        atomicAdd(o + 0, v.x);
        atomicAdd(o + 1, v.y);
        atomicAdd(o + 2, v.z);
        atomicAdd(o + 3, v.w);
    }
}

// ---------------------------------------------------------------------------
// out = opt_relu(agg * norm_dst[row] + bias[col] (+ res))
// ---------------------------------------------------------------------------
template <bool HAS_RES, bool RELU>
__global__ void finish_kernel(const float* __restrict__ agg,
                              const float* __restrict__ nd,
                              const float* __restrict__ bias,
                              const float* __restrict__ res,
                              float* __restrict__ out,
                              int total) {
    int t = blockIdx.x * blockDim.x + threadIdx.x;
    if (t < total) {
        int i = t >> 7;          // / NFEAT
        int f = t & (NFEAT - 1); // % NFEAT
        float v = agg[t] * nd[i] + bias[f];
        if (HAS_RES) v += res[t];
        if (RELU) v = fmaxf(v, 0.0f);
        out[t] = v;
    }
}

extern "C" void kernel_launch(void* const* d_in, const int* in_sizes, int n_in,
                              void* d_out, int out_size, void* d_ws, size_t ws_size,
                              hipStream_t stream) {
    const float* x         = (const float*)d_in[0];
    const long long* ei    = (const long long*)d_in[1];
    const float* W_res     = (const float*)d_in[2];
    const float* b_res     = (const float*)d_in[3];
    const float* W1        = (const float*)d_in[4];
    const float* b1        = (const float*)d_in[5];
    const float* W2        = (const float*)d_in[6];
    const float* b2        = (const float*)d_in[7];
    const float* W_op      = (const float*)d_in[8];
    const float* b_op      = (const float*)d_in[9];

    const int N = in_sizes[0] / NFEAT;   // 50000
    const int E = in_sizes[1] / 2;       // 800000
    const long long* src = ei;
    const long long* dst = ei + E;

    // Workspace layout
    float* ws       = (float*)d_ws;
    float* norm_src = ws;                 ws += N;
    float* norm_dst = ws;                 ws += N;
    float* buf_res  = ws;                 ws += (size_t)N * NFEAT;
    float* buf_h    = ws;                 ws += (size_t)N * NFEAT;
    float* buf_agg  = ws;                 ws += (size_t)N * NFEAT;

    const size_t featBytes = (size_t)N * NFEAT * sizeof(float);
    const int thr = 256;
    const int gemmGrid = (N + 31) / 32;

    // 1) degrees -> norms (norm_src/norm_dst contiguous: one memset, one pass)
    hipMemsetAsync(norm_src, 0, (size_t)2 * N * sizeof(float), stream);
    deg_kernel<<<(E + thr - 1) / thr, thr, 0, stream>>>(src, dst, norm_src, norm_dst, E);
    norm_kernel<<<(2 * N + thr - 1) / thr, thr, 0, stream>>>(norm_src, 2 * N);

    // 2) res = x @ W_res + b_res
    gemm_wmma_f32<NFEAT, NFEAT, false, true, false>
        <<<gemmGrid, 32 * (NFEAT / 32), 0, stream>>>(x, W_res, b_res, nullptr, buf_res, N);

    // 3) h1 = (x * norm_src) @ W1   (bias applied post-aggregation)
    gemm_wmma_f32<NFEAT, NFEAT, true, false, false>
        <<<gemmGrid, 32 * (NFEAT / 32), 0, stream>>>(x, W1, nullptr, norm_src, buf_h, N);

    // 4) agg1 = segment_sum(h1[src], dst)
    hipMemsetAsync(buf_agg, 0, featBytes, stream);
    {
        long long nthreads = (long long)E * 32;
        scatter_kernel<<<(unsigned)((nthreads + thr - 1) / thr), thr, 0, stream>>>(
            buf_h, src, dst, buf_agg, E);
    }

    // 5) out1 = relu(agg1 * norm_dst + b1)   -> buf_h
    finish_kernel<false, true><<<(N * NFEAT + thr - 1) / thr, thr, 0, stream>>>(
        buf_agg, norm_dst, b1, nullptr, buf_h, N * NFEAT);

    // 6) h2 = (out1 * norm_src) @ W2   -> buf_agg (agg1 dead)
    gemm_wmma_f32<NFEAT, NFEAT, true, false, false>
        <<<gemmGrid, 32 * (NFEAT / 32), 0, stream>>>(buf_h, W2, nullptr, norm_src, buf_agg, N);

    // 7) agg2 = segment_sum(h2[src], dst)  -> buf_h (out1 dead)
    hipMemsetAsync(buf_h, 0, featBytes, stream);
    {
        long long nthreads = (long long)E * 32;
        scatter_kernel<<<(unsigned)((nthreads + thr - 1) / thr), thr, 0, stream>>>(
            buf_agg, src, dst, buf_h, E);
    }

    // 8) out2 = relu(agg2 * norm_dst + b2 + res)  -> buf_res (elementwise, in-place OK)
    finish_kernel<true, true><<<(N * NFEAT + thr - 1) / thr, thr, 0, stream>>>(
        buf_h, norm_dst, b2, buf_res, buf_res, N * NFEAT);

    // 9) final = out2 @ W_op + b_op  -> d_out  (FOUT = 64 -> 2 waves/block)
    gemm_wmma_f32<NFEAT, 64, false, true, false>
        <<<gemmGrid, 32 * (64 / 32), 0, stream>>>(buf_res, W_op, b_op, nullptr, (float*)d_out, N);
}